// CapsuleLayer_34273839022349
// MI455X (gfx1250) — compile-verified
//
#include <hip/hip_runtime.h>
#include <hip/hip_bf16.h>

typedef __attribute__((ext_vector_type(2))) float v2f;
typedef __attribute__((ext_vector_type(4))) float v4f;
typedef __attribute__((ext_vector_type(8))) float v8f;

#define NB 32      // batch
#define NI 2048    // input capsules
#define DK 16      // input capsule dim (K of GEMM)
#define NJ 32      // output capsules
#define DD 32      // output capsule dim

constexpr int BLOCKS = 256;
constexpr int STRIP  = NI / BLOCKS;     // 8 i-values per block

// ---------------------------------------------------------------- zero
__global__ void zero_kernel(float* __restrict__ p, int n) {
    int id = blockIdx.x * blockDim.x + threadIdx.x;
    if (id < n) p[id] = 0.0f;
}

// ---------------------------------------------------------------- routing pass
// R = routing iteration (0,1,2). Recomputes u_hat via WMMA f32 16x16x4,
// does agreement + softmax + c-weighted partial sum in one sweep over I.
template <int R>
__global__ __launch_bounds__(256)
void routing_pass(const float* __restrict__ inp,      // [B][I][DK]
                  const float* __restrict__ W,        // [J][I][DD][DK]
                  const float* __restrict__ out_t,    // prev outputs, tile layout
                  float* __restrict__ b_ws,           // logits [I][B][J]
                  float* __restrict__ s_accum)        // [B][J][DD]
{
    extern __shared__ float smem[];
    float* u_lds = smem;                    // 32*4*32*8 = 32768 f (R>0 only)
    float* l_lds = smem + 32768;            // logits [b][j] : 1024 f
    float* c_lds = smem + 32768 + 1024;     // softmax [b][j]: 1024 f

    const int tid    = threadIdx.x;
    const int wave   = tid >> 5;
    const int L      = tid & 31;
    const int lane16 = L & 15;
    const int half   = L >> 4;
    const int j0     = wave * 4;

    // per-wave partial s accumulators: s[b, j(wave-owned), d]
    v8f s_acc[4][2][2];
    #pragma unroll
    for (int a = 0; a < 4; ++a)
        #pragma unroll
        for (int m = 0; m < 2; ++m)
            #pragma unroll
            for (int n = 0; n < 2; ++n)
                #pragma unroll
                for (int e = 0; e < 8; ++e) s_acc[a][m][n][e] = 0.0f;

    for (int ii = 0; ii < STRIP; ++ii) {
        const int i = blockIdx.x * STRIP + ii;

        // ---- A operand: inputs[b, i, k] in WMMA 16x4 layout (2 M-tiles, 4 k-steps)
        v2f a_reg[2][4];
        #pragma unroll
        for (int mt = 0; mt < 2; ++mt)
            #pragma unroll
            for (int s = 0; s < 4; ++s) {
                const int b = mt * 16 + lane16;
                const int k = 4 * s + 2 * half;
                a_reg[mt][s] = *(const v2f*)&inp[((size_t)b * NI + i) * DK + k];
            }

        // warm L2 for next i's W slab (global_prefetch_b8)
        if (ii + 1 < STRIP)
            __builtin_prefetch(&W[(((size_t)j0 * NI + (i + 1)) * DD) * DK], 0, 1);

        #pragma unroll
        for (int jj = 0; jj < 4; ++jj) {
            const int j = j0 + jj;
            const float* Wji = &W[(((size_t)j * NI + i) * DD) * DK];

            // ---- B operand: W[j,i,d,k] already in WMMA KxN layout per-lane
            v2f b_reg[2][4];
            #pragma unroll
            for (int nt = 0; nt < 2; ++nt)
                #pragma unroll
                for (int s = 0; s < 4; ++s) {
                    const int d = nt * 16 + lane16;
                    const int k = 4 * s + 2 * half;
                    b_reg[nt][s] = *(const v2f*)&Wji[d * DK + k];
                }

            // ---- u_hat tile: 4 WMMA-accumulated k-steps per 16x16 tile
            v8f acc[2][2];
            #pragma unroll
            for (int mt = 0; mt < 2; ++mt)
                #pragma unroll
                for (int nt = 0; nt < 2; ++nt) {
                    v8f c;
                    #pragma unroll
                    for (int e = 0; e < 8; ++e) c[e] = 0.0f;
                    #pragma unroll
                    for (int s = 0; s < 4; ++s)
                        c = __builtin_amdgcn_wmma_f32_16x16x4_f32(
                                false, a_reg[mt][s], false, b_reg[nt][s],
                                (short)0, c, false, false);
                    acc[mt][nt] = c;
                }

            if constexpr (R == 0) {
                // softmax(0) == 1/32: accumulate directly
                #pragma unroll
                for (int mt = 0; mt < 2; ++mt)
                    #pragma unroll
                    for (int nt = 0; nt < 2; ++nt)
                        s_acc[jj][mt][nt] += acc[mt][nt] * (1.0f / 32.0f);
            } else {
                // stage u_hat tiles in LDS; agreement a[b,j] = sum_d out_prev*u_hat
                v8f agg[2];
                #pragma unroll
                for (int e = 0; e < 8; ++e) { agg[0][e] = 0.0f; agg[1][e] = 0.0f; }

                #pragma unroll
                for (int mt = 0; mt < 2; ++mt)
                    #pragma unroll
                    for (int nt = 0; nt < 2; ++nt) {
                        const int base = ((j * 4 + mt * 2 + nt) * 32 + L) * 8;
                        v4f lo = __builtin_shufflevector(acc[mt][nt], acc[mt][nt], 0, 1, 2, 3);
                        v4f hi = __builtin_shufflevector(acc[mt][nt], acc[mt][nt], 4, 5, 6, 7);
                        *(v4f*)&u_lds[base]     = lo;
                        *(v4f*)&u_lds[base + 4] = hi;
                        v8f o = *(const v8f*)&out_t[base];   // prev outputs, same tile layout
                        agg[mt] += acc[mt][nt] * o;
                    }

                // reduce over d (lanes within each 16-half), keep b = mt*16+v+8*half
                #pragma unroll
                for (int mt = 0; mt < 2; ++mt) {
                    v8f r = agg[mt];
                    #pragma unroll
                    for (int off = 1; off < 16; off <<= 1) {
                        v8f t;
                        #pragma unroll
                        for (int e = 0; e < 8; ++e) t[e] = __shfl_xor(r[e], off, 32);
                        r += t;
                    }
                    if (lane16 == 0) {
                        #pragma unroll
                        for (int v = 0; v < 8; ++v) {
                            const int b = mt * 16 + v + 8 * half;
                            l_lds[b * 32 + j] = r[v];
                        }
                    }
                }
            }
        } // jj

        if constexpr (R != 0) {
            __syncthreads();
            // softmax over J per batch row; thread t handles b = t
            if (tid < 32) {
                const int b = tid;
                float l[NJ];
                float m = -1e30f;
                #pragma unroll
                for (int j = 0; j < NJ; ++j) {
                    float v = l_lds[b * 32 + j];
                    if constexpr (R == 2) v += b_ws[(size_t)i * 1024 + b * 32 + j];
                    if constexpr (R == 1) b_ws[(size_t)i * 1024 + b * 32 + j] = v;
                    l[j] = v;
                    m = fmaxf(m, v);
                }
                float sum = 0.0f;
                #pragma unroll
                for (int j = 0; j < NJ; ++j) { l[j] = __expf(l[j] - m); sum += l[j]; }
                const float inv = 1.0f / sum;
                #pragma unroll
                for (int j = 0; j < NJ; ++j) c_lds[b * 32 + j] = l[j] * inv;
            }
            __syncthreads();

            // c-weighted accumulation from staged u_hat tiles
            #pragma unroll
            for (int jj = 0; jj < 4; ++jj) {
                const int j = j0 + jj;
                #pragma unroll
                for (int mt = 0; mt < 2; ++mt) {
                    v8f c;
                    #pragma unroll
                    for (int v = 0; v < 8; ++v)
                        c[v] = c_lds[(mt * 16 + v + 8 * half) * 32 + j];   // LDS broadcast
                    #pragma unroll
                    for (int nt = 0; nt < 2; ++nt) {
                        const int base = ((j * 4 + mt * 2 + nt) * 32 + L) * 8;
                        v8f u = *(const v8f*)&u_lds[base];
                        s_acc[jj][mt][nt] += c * u;
                    }
                }
            }
            __syncthreads();   // u_lds reused next i
        }
    } // i strip

    // flush partial sums (wave-exclusive j, cross-block via fp32 HW atomics)
    #pragma unroll
    for (int jj = 0; jj < 4; ++jj) {
        const int j = j0 + jj;
        #pragma unroll
        for (int mt = 0; mt < 2; ++mt)
            #pragma unroll
            for (int nt = 0; nt < 2; ++nt)
                #pragma unroll
                for (int v = 0; v < 8; ++v) {
                    const int b = mt * 16 + v + 8 * half;
                    const int d = nt * 16 + lane16;
                    unsafeAtomicAdd(&s_accum[((size_t)b * 32 + j) * 32 + d],
                                    s_acc[jj][mt][nt][v]);
                }
    }
}

// ---------------------------------------------------------------- squash
// out = s/sqrt(sn+eps) * sn/(1+sn). Writes either WMMA-tile layout (for the
// next pass's agreement loads) or the natural [B,J,D] final output.
template <bool FINAL>
__global__ void squash_kernel(const float* __restrict__ s_accum,
                              float* __restrict__ out_t,
                              float* __restrict__ dout)
{
    const int t = blockIdx.x * blockDim.x + threadIdx.x;   // 0..1023 = b*32+j
    if (t >= NB * NJ) return;
    const int b = t >> 5, j = t & 31;
    const float* s = &s_accum[(size_t)t * DD];
    float v[DD];
    float sn = 0.0f;
    #pragma unroll
    for (int d = 0; d < DD; ++d) { v[d] = s[d]; sn += v[d] * v[d]; }
    const float scale = (sn / (1.0f + sn)) * rsqrtf(sn + 1e-7f);
    #pragma unroll
    for (int d = 0; d < DD; ++d) {
        const float o = v[d] * scale;
        if (FINAL) {
            dout[(size_t)t * DD + d] = o;
        } else {
            const int mt = b >> 4, v8 = b & 7, hf = (b >> 3) & 1;
            const int nt = d >> 4, ln = d & 15;
            out_t[(((j * 4 + mt * 2 + nt) * 32) + hf * 16 + ln) * 8 + v8] = o;
        }
    }
}

// ---------------------------------------------------------------- launch
extern "C" void kernel_launch(void* const* d_in, const int* in_sizes, int n_in,
                              void* d_out, int out_size, void* d_ws, size_t ws_size,
                              hipStream_t stream) {
    const float* inp = (const float*)d_in[0];   // [32,2048,16]
    const float* W   = (const float*)d_in[1];   // [32,2048,32,16]

    float* s_accum = (float*)d_ws;              // 32768 f
    float* out_t   = s_accum + 32768;           // 32768 f
    float* b_ws    = out_t + 32768;             // 2*1024*1024 f (8 MB)

    const size_t SMEM = (32768 + 1024 + 1024) * sizeof(float);  // 136 KB

    zero_kernel<<<128, 256, 0, stream>>>(s_accum, NB * NJ * DD);
    routing_pass<0><<<BLOCKS, 256, 0, stream>>>(inp, W, out_t, b_ws, s_accum);
    squash_kernel<false><<<4, 256, 0, stream>>>(s_accum, out_t, nullptr);

    zero_kernel<<<128, 256, 0, stream>>>(s_accum, NB * NJ * DD);
    routing_pass<1><<<BLOCKS, 256, SMEM, stream>>>(inp, W, out_t, b_ws, s_accum);
    squash_kernel<false><<<4, 256, 0, stream>>>(s_accum, out_t, nullptr);

    zero_kernel<<<128, 256, 0, stream>>>(s_accum, NB * NJ * DD);
    routing_pass<2><<<BLOCKS, 256, SMEM, stream>>>(inp, W, out_t, b_ws, s_accum);
    squash_kernel<true><<<4, 256, 0, stream>>>(s_accum, out_t, (float*)d_out);
}